// DiscreteMixture_30219389895279
// MI455X (gfx1250) — compile-verified
//
#include <hip/hip_runtime.h>
#include <stdint.h>

namespace {
constexpr int K = 64;
constexpr int D = 2048;
constexpr int TWO_D = 2 * D;                           // 4096 floats = 16 KB
constexpr long long STRIDE = K + (long long)K * TWO_D; // 262208 floats per params row
constexpr int BLOCK = 256;                             // 8 wave32
constexpr int OUT_F4 = D / 4;                          // 512 float4 outputs per row
constexpr int ITERS = OUT_F4 / BLOCK;                  // 2 float4 per thread
constexpr int COPY_ITERS = (TWO_D / 4) / BLOCK;        // 4 async b128 per thread
}

__global__ __launch_bounds__(BLOCK)
void discrete_mixture_kernel(const float* __restrict__ params,
                             const float* __restrict__ noise,
                             const float* __restrict__ eps,
                             float* __restrict__ out)
{
    __shared__ __align__(16) float s_comp[TWO_D];   // staged component row (16 KB)
    __shared__ unsigned long long s_key;

    const int b = blockIdx.x;
    const int t = threadIdx.x;
    const float* __restrict__ prow = params + (size_t)b * (size_t)STRIDE;

    // ---- Preload eps into registers; LOADcnt latency overlaps the argmax phase.
    const float4* __restrict__ eps4 = (const float4*)(eps + (size_t)b * D);
    float4 e[ITERS];
    #pragma unroll
    for (int i = 0; i < ITERS; ++i) e[i] = eps4[t + i * BLOCK];

    if (t == 0) s_key = 0ull;
    __syncthreads();

    // ---- Phase 1: Gumbel-max argmax over K=64 logits.
    // Order-preserving float->uint key, low bits (K-1-k) so equal values
    // resolve to the smallest k (matches jnp.argmax first-index semantics).
    if (t < K) {
        float u = noise[b * K + t];
        float g = -logf(-logf(u));                 // Gumbel noise
        float v = prow[t] + g;
        uint32_t bits = __float_as_uint(v);
        uint32_t ord  = (bits & 0x80000000u) ? ~bits : (bits | 0x80000000u);
        unsigned long long key =
            ((unsigned long long)ord << 32) | (unsigned long long)(K - 1 - t);
        atomicMax(&s_key, key);                    // ds_max_u64 (wave32-safe)
    }
    __syncthreads();

    const int selected = (K - 1) - (int)(s_key & 0xFFFFFFFFull);
    const float* __restrict__ comp = prow + K + (size_t)selected * TWO_D;

    // ---- Phase 2: async DMA the gathered 16 KB row (mean | log_std) into LDS.
    // comp is workgroup-uniform -> SGPR-pair saddr + per-lane 32-bit offset.
    const uint32_t lds_base = (uint32_t)(uintptr_t)(&s_comp[0]);
    #pragma unroll
    for (int i = 0; i < COPY_ITERS; ++i) {
        uint32_t off = (uint32_t)(t + i * BLOCK) * 16u;
        asm volatile("global_load_async_to_lds_b128 %0, %1, %2"
                     :: "v"(lds_base + off), "v"(off), "s"(comp)
                     : "memory");
    }
    asm volatile("s_wait_asynccnt 0" ::: "memory");
    __syncthreads();   // other waves' async writes visible after their own waits

    // ---- Phase 3: out = mean + exp(log_std) * eps, 128-bit in/out.
    float4* __restrict__ out4 = (float4*)(out + (size_t)b * D);
    const float4* __restrict__ s4 = (const float4*)s_comp;
    #pragma unroll
    for (int i = 0; i < ITERS; ++i) {
        int c = t + i * BLOCK;
        float4 m  = s4[c];
        float4 ls = s4[c + OUT_F4];   // log_std lives D floats after mean
        float4 r;
        r.x = fmaf(expf(ls.x), e[i].x, m.x);
        r.y = fmaf(expf(ls.y), e[i].y, m.y);
        r.z = fmaf(expf(ls.z), e[i].z, m.z);
        r.w = fmaf(expf(ls.w), e[i].w, m.w);
        out4[c] = r;
    }
}

extern "C" void kernel_launch(void* const* d_in, const int* in_sizes, int n_in,
                              void* d_out, int out_size, void* d_ws, size_t ws_size,
                              hipStream_t stream) {
    const float* params = (const float*)d_in[0];   // (B, K + K*2D) fp32
    const float* noise  = (const float*)d_in[1];   // (B, K) fp32
    const float* eps    = (const float*)d_in[2];   // (B, D) fp32
    float* out = (float*)d_out;                    // (B, D) fp32

    const int B = in_sizes[1] / K;                 // uniform_noise is (B, K)
    discrete_mixture_kernel<<<B, BLOCK, 0, stream>>>(params, noise, eps, out);
}